// MultiBoxLoss_70300024701370
// MI455X (gfx1250) — compile-verified
//
#include <hip/hip_runtime.h>
#include <hip/hip_bf16.h>
#include <math.h>

#define B_ 128
#define P_ 8732
#define C_ 21
#define O_ 32
#define THRESH 0.5f
#define NEGPOS 3

typedef float v2f __attribute__((ext_vector_type(2)));
typedef float v8f __attribute__((ext_vector_type(8)));
typedef unsigned int u32x4 __attribute__((ext_vector_type(4)));
typedef int i32x8 __attribute__((ext_vector_type(8)));
typedef int i32x4 __attribute__((ext_vector_type(4)));

// Wave32 sum using V_WMMA_F32_16X16X4_F32 (A = lane values in 16x4 layout,
// B = ones -> D[m,n] = v_m + v_{m+16}; 8 adds + one xor-16 shuffle finish).
__device__ __forceinline__ float wave_sum_wmma(float v) {
  v2f a; a[0] = v;    a[1] = 0.0f;
  v2f b; b[0] = 1.0f; b[1] = 1.0f;
  v8f c = {};
  v8f d = __builtin_amdgcn_wmma_f32_16x16x4_f32(false, a, false, b,
                                                (short)0, c, false, false);
  float t = d[0] + d[1] + d[2] + d[3] + d[4] + d[5] + d[6] + d[7];
  t += __shfl_xor(t, 16, 32);
  return t;
}

// TDM: 1-D bulk copy of `nelem` f32 elements from global to LDS.
// D# per CDNA5 ISA (group0: count/lds/global/type,
// group1: data_size=4B, tensor_dim0=nelem, tensor_dim1=1, tile_dim0=nelem,
// tile_dim1=1, tensor_dim0_stride=nelem). Groups 2/3 unused (<=2D tensor).
// This toolchain's builtin is the 6-arg form:
// (u32x4 g0, i32x8 g1, i32x4 g2, i32x4 g3, i32x8, i32 cpol).
__device__ __forceinline__ void tdm_load_f32_1d(const float* gsrc,
                                                unsigned lds_byte_addr,
                                                unsigned nelem) {
  unsigned long long ga = (unsigned long long)(uintptr_t)gsrc;
  u32x4 g0;
  g0[0] = 1u;                                        // count=1, user D#
  g0[1] = lds_byte_addr;                             // lds_addr (bytes)
  g0[2] = (unsigned)(ga & 0xFFFFFFFFull);            // global_addr[31:0]
  g0[3] = (unsigned)((ga >> 32) & 0x01FFFFFFull)     // global_addr[56:32]
        | (2u << 30);                                // type=2 ("image")
  i32x8 g1;
  g1[0] = (int)(2u << 16);                           // data_size=2 (4B); mask=0
  g1[1] = (int)((nelem & 0xFFFFu) << 16);            // tensor_dim0[15:0]
  g1[2] = (int)(((nelem >> 16) & 0xFFFFu)            // tensor_dim0[31:16]
        |       (1u << 16));                         // tensor_dim1 = 1
  g1[3] = (int)((nelem & 0xFFFFu) << 16);            // tile_dim0 = nelem
  g1[4] = 1;                                         // tile_dim1=1, tile_dim2=0
  g1[5] = (int)nelem;                                // tensor_dim0_stride[31:0]
  g1[6] = 0;                                         // stride[47:32]=0
  g1[7] = 0;
  i32x4 z4 = {0, 0, 0, 0};
  i32x8 z8 = {0, 0, 0, 0, 0, 0, 0, 0};
  __builtin_amdgcn_tensor_load_to_lds(g0, g1, z4, z4, z8, 0);
}

// One block per batch image. All intermediates + the full priors tensor live
// in LDS (CDNA5: 320KB/WGP; total here ~220KB).
__global__ __launch_bounds__(1024) void mbl_match_kernel(
    const float* __restrict__ ploc,    // [B,P,4]
    const float* __restrict__ pscore,  // [B,P,C]
    const float* __restrict__ boxes,   // [B,O,4] xy
    const int*   __restrict__ labels,  // [B,O]
    const float* __restrict__ priors,  // [P,4] cxcy
    float* __restrict__ part)          // [4*B]: n_pos, pos_conf, loc_sum, hard
{
  const int b    = blockIdx.x;
  const int tid  = threadIdx.x;
  const int nthr = blockDim.x;
  const int lane = tid & 31;
  const int wave = tid >> 5;
  const int nwaves = nthr >> 5;

  __shared__ float4 sh_pri[P_];             // priors (cxcy), TDM-staged
  __shared__ float sh_ov[P_];               // best overlap per prior
  __shared__ float sh_neg[P_];              // negative CE loss per prior
  __shared__ unsigned char sh_obj[P_];      // best object per prior
  __shared__ float4 sh_box[O_];             // object boxes (xy)
  __shared__ float  sh_barea[O_];           // object areas
  __shared__ int    sh_lab[O_];             // object labels
  __shared__ unsigned long long sh_best[O_];// packed per-object argmax
  __shared__ float sh_part[32][4];          // per-wave partials
  __shared__ float sh_red[8];               // block-level scalars

  // ---- Phase 0: kick off TDM bulk load of priors into LDS (wave 0 only:
  // tensor ops are per-wave and ignore EXEC; one wave => one DMA) ----
  if (tid < 32) {
    tdm_load_f32_1d(priors, (unsigned)(uintptr_t)&sh_pri[0], (unsigned)(P_ * 4));
  }

  // ---- Phase 1: objects to LDS (overlaps with the TDM transfer) ----
  if (tid < O_) {
    float4 bx = reinterpret_cast<const float4*>(boxes)[b * O_ + tid];
    sh_box[tid]   = bx;
    sh_barea[tid] = (bx.z - bx.x) * (bx.w - bx.y);
    sh_lab[tid]   = labels[b * O_ + tid];
    sh_best[tid]  = 0xFFFFFFFFull;  // iou=0 bits, prior index 0
  }
  if (tid < 32) {
    __builtin_amdgcn_s_wait_tensorcnt(0);   // wave 0: TDM done
  }
  __syncthreads();                          // publish LDS priors to all waves

  // ---- Phase 2: IoU scan over priors (LDS-resident) ----
  float    lmax[O_];
  unsigned lp[O_];
#pragma unroll
  for (int o = 0; o < O_; ++o) { lmax[o] = -1.0f; lp[o] = 0u; }

  for (int p = tid; p < P_; p += nthr) {
    float4 pr = sh_pri[p];
    float px0 = pr.x - pr.z * 0.5f, py0 = pr.y - pr.w * 0.5f;
    float px1 = pr.x + pr.z * 0.5f, py1 = pr.y + pr.w * 0.5f;
    float parea = pr.z * pr.w;
    float best = -1.0f; int besto = 0;
#pragma unroll
    for (int o = 0; o < O_; ++o) {
      float4 bx = sh_box[o];
      float ix = fminf(bx.z, px1) - fmaxf(bx.x, px0);
      float iy = fminf(bx.w, py1) - fmaxf(bx.y, py0);
      float inter = fmaxf(ix, 0.0f) * fmaxf(iy, 0.0f);
      float iou = inter / (sh_barea[o] + parea - inter);
      if (iou > best)    { best = iou;    besto = o; }           // first-max wins
      if (iou > lmax[o]) { lmax[o] = iou; lp[o] = (unsigned)p; } // first-max wins
    }
    sh_ov[p]  = best;
    sh_obj[p] = (unsigned char)besto;
  }
#pragma unroll
  for (int o = 0; o < O_; ++o) {
    if (lmax[o] >= 0.0f) {
      unsigned long long key =
          ((unsigned long long)__float_as_uint(lmax[o]) << 32) |
          (unsigned long long)(0xFFFFFFFFu - lp[o]);   // ~p: smaller p wins ties
      atomicMax(&sh_best[o], key);
    }
  }
  __syncthreads();

  // ---- Phase 3: forced matching (sequential scatter, last o wins) ----
  if (tid == 0) {
    for (int o = 0; o < O_; ++o) {
      unsigned pstar = 0xFFFFFFFFu - (unsigned)(sh_best[o] & 0xFFFFFFFFull);
      sh_ov[pstar]  = 1.0f;
      sh_obj[pstar] = (unsigned char)o;
    }
  }
  __syncthreads();

  // ---- Phase 4: single pass over scores/locs ----
  float acc_np = 0.0f, acc_pos = 0.0f, acc_loc = 0.0f;
  for (int p = tid; p < P_; p += nthr) {
    int   o  = (int)sh_obj[p];
    float ov = sh_ov[p];
    int   lab = (ov < THRESH) ? 0 : sh_lab[o];

    const float* sc = pscore + ((size_t)b * P_ + p) * C_;
    __builtin_prefetch(sc + (size_t)nthr * C_, 0, 1);  // global_prefetch_b8
    float x[C_];
    float m = -3.402823466e+38f;
#pragma unroll
    for (int cc = 0; cc < C_; ++cc) { x[cc] = sc[cc]; m = fmaxf(m, x[cc]); }
    float s = 0.0f;
#pragma unroll
    for (int cc = 0; cc < C_; ++cc) s += __expf(x[cc] - m);
    float ce = (m + __logf(s)) - sc[lab];

    if (lab != 0) {
      acc_np  += 1.0f;
      acc_pos += ce;
      sh_neg[p] = 0.0f;
      float4 bx = sh_box[o];
      float bcx = (bx.x + bx.z) * 0.5f, bcy = (bx.y + bx.w) * 0.5f;
      float bw = bx.z - bx.x, bh = bx.w - bx.y;
      float4 pr = sh_pri[p];
      float g0 = (bcx - pr.x) / (pr.z * 0.1f);
      float g1 = (bcy - pr.y) / (pr.w * 0.1f);
      float g2 = __logf(bw / pr.z) * 5.0f;
      float g3 = __logf(bh / pr.w) * 5.0f;
      float4 pl = reinterpret_cast<const float4*>(ploc)[(size_t)b * P_ + p];
      acc_loc += fabsf(pl.x - g0) + fabsf(pl.y - g1) +
                 fabsf(pl.z - g2) + fabsf(pl.w - g3);
    } else {
      sh_neg[p] = ce;
    }
  }

  // deterministic block reduce: WMMA wave sums -> LDS -> serial combine
  float wnp  = wave_sum_wmma(acc_np);
  float wpos = wave_sum_wmma(acc_pos);
  float wloc = wave_sum_wmma(acc_loc);
  if (lane == 0) {
    sh_part[wave][0] = wnp; sh_part[wave][1] = wpos; sh_part[wave][2] = wloc;
  }
  __syncthreads();
  if (tid == 0) {
    float a = 0, bb = 0, cgl = 0;
    for (int w = 0; w < nwaves; ++w) {
      a += sh_part[w][0]; bb += sh_part[w][1]; cgl += sh_part[w][2];
    }
    sh_red[0] = a; sh_red[1] = bb; sh_red[2] = cgl;
  }
  __syncthreads();

  // ---- Phase 5: exact top-k sum of negatives via float-bit bisection ----
  int n_pos = (int)sh_red[0];
  long long k = (long long)NEGPOS * (long long)n_pos;
  if (k > P_) k = P_;

  if (k > 0) {
    // invariant: count(bits>=lo) >= k, count(bits>=hi) < k (values finite >= 0)
    unsigned lo = 0u, hi = 0x7F800001u;
    while (hi - lo > 1u) {
      unsigned mid = lo + ((hi - lo) >> 1);
      float lc = 0.0f;
      for (int p = tid; p < P_; p += nthr)
        if (__float_as_uint(sh_neg[p]) >= mid) lc += 1.0f;
      float wc = wave_sum_wmma(lc);
      if (lane == 0) sh_part[wave][0] = wc;
      __syncthreads();
      if (tid == 0) {
        float s2 = 0;
        for (int w = 0; w < nwaves; ++w) s2 += sh_part[w][0];
        sh_red[3] = s2;
      }
      __syncthreads();
      float cnt = sh_red[3];                 // exact integer in f32 (<= 8732)
      if ((long long)cnt >= k) lo = mid; else hi = mid;  // uniform branch
    }
    float kth = __uint_as_float(lo);         // exact k-th largest value
    float lsum = 0.0f, lcg = 0.0f;
    for (int p = tid; p < P_; p += nthr) {
      float v = sh_neg[p];
      if (__float_as_uint(v) > lo) { lsum += v; lcg += 1.0f; }
    }
    float wsum = wave_sum_wmma(lsum);
    float wcg  = wave_sum_wmma(lcg);
    if (lane == 0) { sh_part[wave][0] = wsum; sh_part[wave][1] = wcg; }
    __syncthreads();
    if (tid == 0) {
      float s2 = 0, c2 = 0;
      for (int w = 0; w < nwaves; ++w) { s2 += sh_part[w][0]; c2 += sh_part[w][1]; }
      part[3 * B_ + b] = s2 + ((float)k - c2) * kth;  // ties land on kth value
    }
  } else if (tid == 0) {
    part[3 * B_ + b] = 0.0f;
  }

  if (tid == 0) {
    part[0 * B_ + b] = sh_red[0];
    part[1 * B_ + b] = sh_red[1];
    part[2 * B_ + b] = sh_red[2];
  }
}

// Single-wave final combine (WMMA reduction again).
__global__ __launch_bounds__(32) void mbl_final_kernel(
    const float* __restrict__ part, float* __restrict__ out)
{
  int lane = threadIdx.x;
  float np = 0.0f, pos = 0.0f, loc = 0.0f, hard = 0.0f;
  for (int b = lane; b < B_; b += 32) {
    np   += part[0 * B_ + b];
    pos  += part[1 * B_ + b];
    loc  += part[2 * B_ + b];
    hard += part[3 * B_ + b];
  }
  np   = wave_sum_wmma(np);
  pos  = wave_sum_wmma(pos);
  loc  = wave_sum_wmma(loc);
  hard = wave_sum_wmma(hard);
  if (lane == 0) {
    float conf_loss = (hard + pos) / np;
    float loc_loss  = loc / (np * 4.0f);
    out[0] = conf_loss + loc_loss;
  }
}

extern "C" void kernel_launch(void* const* d_in, const int* in_sizes, int n_in,
                              void* d_out, int out_size, void* d_ws, size_t ws_size,
                              hipStream_t stream) {
  const float* ploc   = (const float*)d_in[0];  // [B,P,4]
  const float* pscore = (const float*)d_in[1];  // [B,P,C]
  const float* boxes  = (const float*)d_in[2];  // [B,O,4]
  const int*   labels = (const int*)d_in[3];    // [B,O]
  const float* priors = (const float*)d_in[4];  // [P,4]
  float* part = (float*)d_ws;                   // 4*B floats

  mbl_match_kernel<<<B_, 1024, 0, stream>>>(ploc, pscore, boxes, labels, priors, part);
  mbl_final_kernel<<<1, 32, 0, stream>>>(part, (float*)d_out);
}